// LoLGATRecommender_2027224563943
// MI455X (gfx1250) — compile-verified
//
#include <hip/hip_runtime.h>
#include <math.h>

#define N_NODES   100000
#define N_EDGES   1600000
#define N_EDGES_SL (N_EDGES + N_NODES)   /* 1,700,000 with self loops */
#define N_GRAPHS  10000
#define EMB_DIM   32
#define HID       64
#define H1        4
#define F1        (H1 * HID)             /* 256 */
#define NEG_SLOPE 0.2f

typedef __attribute__((ext_vector_type(2))) float v2f;
typedef __attribute__((ext_vector_type(8))) float v8f;

// Only check in the DEVICE pass: host pass has no amdgcn builtins declared
// (they are merely tolerated in device-function bodies via aux-target).
#if defined(__HIP_DEVICE_COMPILE__) && !__has_builtin(__builtin_amdgcn_wmma_f32_16x16x4_f32)
#error "gfx1250 f32 WMMA builtin not available in device pass"
#endif

// ---------- ordered-int float max trick (monotonic bijection, involution) ----
__device__ __forceinline__ int f2ord(float f) {
    int i = __float_as_int(f);
    return i >= 0 ? i : (i ^ 0x7FFFFFFF);
}
__device__ __forceinline__ float ord2f(int i) {
    return __int_as_float(i >= 0 ? i : (i ^ 0x7FFFFFFF));
}

// ---------------------------- fills -----------------------------------------
__global__ void k_fill_f32(float* p, float v, int n) {
    int i = blockIdx.x * blockDim.x + threadIdx.x;
    if (i < n) p[i] = v;
}
__global__ void k_fill_i32(int* p, int v, int n) {
    int i = blockIdx.x * blockDim.x + threadIdx.x;
    if (i < n) p[i] = v;
}

// ------------------- layer-1 champion tables (tiny) --------------------------
// g1tab[c][o] = sum_k emb[c][k] * W1[k][o]      (170 x 256)
__global__ void k_table1(const float* __restrict__ emb, const float* __restrict__ W1,
                         float* __restrict__ g1tab) {
    int i = blockIdx.x * blockDim.x + threadIdx.x;
    if (i >= 170 * F1) return;
    int c = i >> 8, o = i & 255;
    float s = 0.f;
#pragma unroll
    for (int k = 0; k < EMB_DIM; ++k) s += emb[c * EMB_DIM + k] * W1[k * F1 + o];
    g1tab[i] = s;
}
// als1[c][h] = g1tab[c][h*64:..] . a_src1[h]   (170 x 4), same for ald1
__global__ void k_table1b(const float* __restrict__ g1tab,
                          const float* __restrict__ a_src1, const float* __restrict__ a_dst1,
                          float* __restrict__ als1, float* __restrict__ ald1) {
    int i = blockIdx.x * blockDim.x + threadIdx.x;
    if (i >= 170 * H1) return;
    int c = i >> 2, h = i & 3;
    float ss = 0.f, sd = 0.f;
#pragma unroll
    for (int k = 0; k < HID; ++k) {
        float g = g1tab[c * F1 + h * HID + k];
        ss += g * a_src1[h * HID + k];
        sd += g * a_dst1[h * HID + k];
    }
    als1[i] = ss;
    ald1[i] = sd;
}

// ------------------- layer-1 edge passes -------------------------------------
__global__ void k_edge1_max(const int* __restrict__ esrc, const int* __restrict__ edst,
                            const int* __restrict__ x,
                            const float* __restrict__ als1, const float* __restrict__ ald1,
                            float* __restrict__ e1, int* __restrict__ max1) {
    int i = blockIdx.x * blockDim.x + threadIdx.x;
    if (i >= N_EDGES_SL) return;
    int s, d;
    if (i < N_EDGES) { s = esrc[i]; d = edst[i]; } else { s = d = i - N_EDGES; }
    int cs = x[s] * H1, cd = x[d] * H1;
#pragma unroll
    for (int h = 0; h < H1; ++h) {
        float e = als1[cs + h] + ald1[cd + h];
        e = (e >= 0.f) ? e : NEG_SLOPE * e;
        e1[i * H1 + h] = e;
        atomicMax(&max1[d * H1 + h], f2ord(e));
    }
}

__global__ void k_edge1_exp(const int* __restrict__ edst,
                            float* __restrict__ e1, const int* __restrict__ max1,
                            float* __restrict__ den1) {
    int i = blockIdx.x * blockDim.x + threadIdx.x;
    if (i >= N_EDGES_SL) return;
    int d = (i < N_EDGES) ? edst[i] : (i - N_EDGES);
#pragma unroll
    for (int h = 0; h < H1; ++h) {
        float e = e1[i * H1 + h];
        float m = ord2f(max1[d * H1 + h]);
        float ex = expf(e - m);
        e1[i * H1 + h] = ex;
        atomicAdd(&den1[d * H1 + h], ex);
    }
}

// one thread = (edge, 4 consecutive channels of 256)
__global__ void k_edge1_agg(const int* __restrict__ esrc, const int* __restrict__ edst,
                            const int* __restrict__ x, const float* __restrict__ g1tab,
                            const float* __restrict__ e1, const float* __restrict__ den1,
                            float* __restrict__ out1) {
    long i = (long)blockIdx.x * blockDim.x + threadIdx.x;
    if (i >= (long)N_EDGES_SL * 64) return;
    int q = (int)(i & 63);
    int ei = (int)(i >> 6);
    int s, d;
    if (ei < N_EDGES) { s = esrc[ei]; d = edst[ei]; } else { s = d = ei - N_EDGES; }
    int h = q >> 4;                                  // 4 channels => head = (4q)/64
    float alpha = e1[ei * H1 + h] / (den1[d * H1 + h] + 1e-16f);
    int cs = x[s];
    float4 g = *(const float4*)(g1tab + cs * F1 + q * 4);
    float* o = out1 + (long)d * F1 + q * 4;
    atomicAdd(o + 0, g.x * alpha);
    atomicAdd(o + 1, g.y * alpha);
    atomicAdd(o + 2, g.z * alpha);
    atomicAdd(o + 3, g.w * alpha);
}

// h2 = elu(out1 + b1), in place
__global__ void k_elu_bias1(float* __restrict__ h2, const float* __restrict__ b1) {
    int i = blockIdx.x * blockDim.x + threadIdx.x;
    if (i >= N_NODES * F1) return;
    float v = h2[i] + b1[i & 255];
    h2[i] = (v > 0.f) ? v : expm1f(v);
}

// ------------------- layer-2 GEMM via f32 WMMA -------------------------------
// g2[N,64] = h2[N,256] @ W2[256,64]; 8 waves/block, each wave owns a 16x64 tile.
__global__ void k_gemm_wmma(const float* __restrict__ h2, const float* __restrict__ W2,
                            float* __restrict__ g2, int nrows) {
    __shared__ float sW[F1 * HID];                   // 64 KB, whole W2
    int tid = threadIdx.x;
    for (int i = tid; i < F1 * HID; i += 256) sW[i] = W2[i];
    __syncthreads();

    int wave = tid >> 5;
    int lane = tid & 31;
    int rowBase = (blockIdx.x * 8 + wave) * 16;
    if (rowBase >= nrows) return;                    // wave-uniform: EXEC all-1s inside

    // A (16x4 f32): lanes 0-15 -> K,K+1 ; lanes 16-31 -> K+2,K+3 ; M = lane&15
    int m    = rowBase + (lane & 15);
    int ksel = (lane >> 4) * 2;
    int col  = lane & 15;
    const float* arow = h2 + (long)m * F1;

    v8f acc[4] = { {}, {}, {}, {} };
    for (int k = 0; k < F1; k += 4) {
        v2f a;
        a.x = arow[k + ksel];
        a.y = arow[k + ksel + 1];
#pragma unroll
        for (int t = 0; t < 4; ++t) {
            v2f b;
            b.x = sW[(k + ksel)     * HID + t * 16 + col];
            b.y = sW[(k + ksel + 1) * HID + t * 16 + col];
            acc[t] = __builtin_amdgcn_wmma_f32_16x16x4_f32(
                false, a, false, b, (short)0, acc[t], false, false);
        }
    }
    // D (16x16 f32): VGPR v holds M=v (lanes 0-15) / M=v+8 (lanes 16-31), N=lane&15
    int rowOff = (lane >> 4) * 8;
#pragma unroll
    for (int t = 0; t < 4; ++t)
#pragma unroll
        for (int v = 0; v < 8; ++v) {
            int r = rowBase + rowOff + v;
            g2[(long)r * HID + t * 16 + col] = acc[t][v];
        }
}

// ------------------- layer-2 attention + edges -------------------------------
__global__ void k_att2(const float* __restrict__ g2,
                       const float* __restrict__ a_src2, const float* __restrict__ a_dst2,
                       float* __restrict__ als2, float* __restrict__ ald2) {
    int n = blockIdx.x * blockDim.x + threadIdx.x;
    if (n >= N_NODES) return;
    float ss = 0.f, sd = 0.f;
#pragma unroll 8
    for (int c = 0; c < HID; ++c) {
        float g = g2[n * HID + c];
        ss += g * a_src2[c];
        sd += g * a_dst2[c];
    }
    als2[n] = ss;
    ald2[n] = sd;
}

__global__ void k_edge2_max(const int* __restrict__ esrc, const int* __restrict__ edst,
                            const float* __restrict__ als2, const float* __restrict__ ald2,
                            float* __restrict__ e2, int* __restrict__ max2) {
    int i = blockIdx.x * blockDim.x + threadIdx.x;
    if (i >= N_EDGES_SL) return;
    int s, d;
    if (i < N_EDGES) { s = esrc[i]; d = edst[i]; } else { s = d = i - N_EDGES; }
    float e = als2[s] + ald2[d];
    e = (e >= 0.f) ? e : NEG_SLOPE * e;
    e2[i] = e;
    atomicMax(&max2[d], f2ord(e));
}

__global__ void k_edge2_exp(const int* __restrict__ edst,
                            float* __restrict__ e2, const int* __restrict__ max2,
                            float* __restrict__ den2) {
    int i = blockIdx.x * blockDim.x + threadIdx.x;
    if (i >= N_EDGES_SL) return;
    int d = (i < N_EDGES) ? edst[i] : (i - N_EDGES);
    float ex = expf(e2[i] - ord2f(max2[d]));
    e2[i] = ex;
    atomicAdd(&den2[d], ex);
}

// one thread = (edge, 4 consecutive channels of 64)
__global__ void k_edge2_agg(const int* __restrict__ esrc, const int* __restrict__ edst,
                            const float* __restrict__ g2,
                            const float* __restrict__ e2, const float* __restrict__ den2,
                            float* __restrict__ out2) {
    long i = (long)blockIdx.x * blockDim.x + threadIdx.x;
    if (i >= (long)N_EDGES_SL * 16) return;
    int q  = (int)(i & 15);
    int ei = (int)(i >> 4);
    int s, d;
    if (ei < N_EDGES) { s = esrc[ei]; d = edst[ei]; } else { s = d = ei - N_EDGES; }
    float alpha = e2[ei] / (den2[d] + 1e-16f);
    float4 g = *(const float4*)(g2 + (long)s * HID + q * 4);
    float* o = out2 + (long)d * HID + q * 4;
    atomicAdd(o + 0, g.x * alpha);
    atomicAdd(o + 1, g.y * alpha);
    atomicAdd(o + 2, g.z * alpha);
    atomicAdd(o + 3, g.w * alpha);
}

// ------------------- pool + head ---------------------------------------------
// pooled@fc_w is linear => per-node scalar s_n = elu(out2[n]+b2) . fc_w, segment-summed
__global__ void k_pool(const float* __restrict__ out2, const float* __restrict__ b2,
                       const float* __restrict__ fcw, const int* __restrict__ batch,
                       float* __restrict__ gsum, float* __restrict__ gcnt) {
    int n = blockIdx.x * blockDim.x + threadIdx.x;
    if (n >= N_NODES) return;
    float s = 0.f;
#pragma unroll 8
    for (int c = 0; c < HID; ++c) {
        float v = out2[n * HID + c] + b2[c];
        v = (v > 0.f) ? v : expm1f(v);
        s += v * fcw[c];
    }
    int g = batch[n];
    atomicAdd(&gsum[g], s);
    atomicAdd(&gcnt[g], 1.f);
}

__global__ void k_head(const float* __restrict__ gsum, const float* __restrict__ gcnt,
                       const float* __restrict__ fcb, float* __restrict__ out) {
    int g = blockIdx.x * blockDim.x + threadIdx.x;
    if (g >= N_GRAPHS) return;
    float z = gsum[g] / fmaxf(gcnt[g], 1.f) + fcb[0];
    out[g] = 1.f / (1.f + expf(-z));
}

// =============================================================================
extern "C" void kernel_launch(void* const* d_in, const int* in_sizes, int n_in,
                              void* d_out, int out_size, void* d_ws, size_t ws_size,
                              hipStream_t stream) {
    const int*   x      = (const int*)d_in[0];
    const int*   eidx   = (const int*)d_in[1];
    const int*   batch  = (const int*)d_in[2];
    const float* emb    = (const float*)d_in[3];
    const float* W1     = (const float*)d_in[4];
    const float* a_src1 = (const float*)d_in[5];
    const float* a_dst1 = (const float*)d_in[6];
    const float* b1     = (const float*)d_in[7];
    const float* W2     = (const float*)d_in[8];
    const float* a_src2 = (const float*)d_in[9];
    const float* a_dst2 = (const float*)d_in[10];
    const float* b2     = (const float*)d_in[11];
    const float* fc_w   = (const float*)d_in[12];
    const float* fc_b   = (const float*)d_in[13];
    float* out = (float*)d_out;
    (void)in_sizes; (void)n_in; (void)out_size;

    const int* esrc = eidx;
    const int* edst = eidx + N_EDGES;

    // ---- carve workspace ----
    char* ws = (char*)d_ws;
    size_t off = 0;
    auto carve = [&](size_t bytes) -> char* {
        char* p = ws + off;
        off += (bytes + 255) & ~(size_t)255;
        return p;
    };
    float* g1tab = (float*)carve((size_t)170 * F1 * 4);
    float* als1  = (float*)carve((size_t)170 * H1 * 4);
    float* ald1  = (float*)carve((size_t)170 * H1 * 4);
    float* h2    = (float*)carve((size_t)N_NODES * F1 * 4);       // out1 -> h2 (in place)
    float* e1    = (float*)carve((size_t)N_EDGES_SL * H1 * 4);    // reused as g2 later
    int*   max1  = (int*)  carve((size_t)N_NODES * H1 * 4);
    float* den1  = (float*)carve((size_t)N_NODES * H1 * 4);
    float* als2  = (float*)carve((size_t)N_NODES * 4);
    float* ald2  = (float*)carve((size_t)N_NODES * 4);
    int*   max2  = (int*)  carve((size_t)N_NODES * 4);
    float* den2  = (float*)carve((size_t)N_NODES * 4);
    float* e2    = (float*)carve((size_t)N_EDGES_SL * 4);
    float* out2  = (float*)carve((size_t)N_NODES * HID * 4);
    float* gsum  = (float*)carve((size_t)N_GRAPHS * 4);
    float* gcnt  = (float*)carve((size_t)N_GRAPHS * 4);
    float* g2    = e1;   // alias: e1 (27.2 MB) no longer needed once g2 (25.6 MB) is built
    if (off > ws_size) return;   // insufficient scratch: no-op (deterministic)

    const int BS = 256;
    const int INT_MINV = (int)0x80000000;
    auto blocks = [](long n, int bs) { return (unsigned)((n + bs - 1) / bs); };

    // ---- init accumulators ----
    k_fill_i32<<<blocks(N_NODES * H1, BS), BS, 0, stream>>>(max1, INT_MINV, N_NODES * H1);
    k_fill_f32<<<blocks(N_NODES * H1, BS), BS, 0, stream>>>(den1, 0.f, N_NODES * H1);
    k_fill_f32<<<blocks((long)N_NODES * F1, BS), BS, 0, stream>>>(h2, 0.f, N_NODES * F1);
    k_fill_i32<<<blocks(N_NODES, BS), BS, 0, stream>>>(max2, INT_MINV, N_NODES);
    k_fill_f32<<<blocks(N_NODES, BS), BS, 0, stream>>>(den2, 0.f, N_NODES);
    k_fill_f32<<<blocks((long)N_NODES * HID, BS), BS, 0, stream>>>(out2, 0.f, N_NODES * HID);
    k_fill_f32<<<blocks(N_GRAPHS, BS), BS, 0, stream>>>(gsum, 0.f, N_GRAPHS);
    k_fill_f32<<<blocks(N_GRAPHS, BS), BS, 0, stream>>>(gcnt, 0.f, N_GRAPHS);

    // ---- layer-1 champion tables ----
    k_table1 <<<blocks(170 * F1, BS), BS, 0, stream>>>(emb, W1, g1tab);
    k_table1b<<<blocks(170 * H1, BS), BS, 0, stream>>>(g1tab, a_src1, a_dst1, als1, ald1);

    // ---- layer-1 edge softmax + aggregation ----
    k_edge1_max<<<blocks(N_EDGES_SL, BS), BS, 0, stream>>>(esrc, edst, x, als1, ald1, e1, max1);
    k_edge1_exp<<<blocks(N_EDGES_SL, BS), BS, 0, stream>>>(edst, e1, max1, den1);
    k_edge1_agg<<<blocks((long)N_EDGES_SL * 64, BS), BS, 0, stream>>>(
        esrc, edst, x, g1tab, e1, den1, h2);
    k_elu_bias1<<<blocks((long)N_NODES * F1, BS), BS, 0, stream>>>(h2, b1);

    // ---- layer-2 GEMM (f32 WMMA) ----
    {
        int rowTiles = N_NODES / 16;                          // 6250
        unsigned gb = (unsigned)((rowTiles + 7) / 8);         // 8 waves/block
        k_gemm_wmma<<<gb, 256, 0, stream>>>(h2, W2, g2, N_NODES);
    }

    // ---- layer-2 attention + edge softmax + aggregation ----
    k_att2<<<blocks(N_NODES, BS), BS, 0, stream>>>(g2, a_src2, a_dst2, als2, ald2);
    k_edge2_max<<<blocks(N_EDGES_SL, BS), BS, 0, stream>>>(esrc, edst, als2, ald2, e2, max2);
    k_edge2_exp<<<blocks(N_EDGES_SL, BS), BS, 0, stream>>>(edst, e2, max2, den2);
    k_edge2_agg<<<blocks((long)N_EDGES_SL * 16, BS), BS, 0, stream>>>(
        esrc, edst, g2, e2, den2, out2);

    // ---- pool + sigmoid head ----
    k_pool<<<blocks(N_NODES, BS), BS, 0, stream>>>(out2, b2, fc_w, batch, gsum, gcnt);
    k_head<<<blocks(N_GRAPHS, BS), BS, 0, stream>>>(gsum, gcnt, fc_b, out);
}